// EfficientMultiTensorProductS2Grid_28217935135341
// MI455X (gfx1250) — compile-verified
//
#include <hip/hip_runtime.h>
#include <math.h>

#define PI_D 3.14159265358979323846

typedef __attribute__((ext_vector_type(2))) float v2f;
typedef __attribute__((ext_vector_type(8))) float v8f;

// =====================================================================
// Workspace layout (floats):
//   [0 .. 1023]   : header, viewed as doubles (GL nodes/weights)
//                   d[0..13]=x(rb=14)  d[16..29]=w(rb=14)
//                   d[40..59]=x(rb=20) d[64..83]=w(rb=20)
//   1024  +3072   : Y1   packed  (K=16 , N=182, KK=4 , 12 tiles)  TP1
//   4096  +11776  : Bo1T packed  (K=182, N=49 , KK=46,  4 tiles)  TP1
//   15872 +19968  : Y1_2 packed  (K=49 , N=380, KK=13, 24 tiles)  TP2
//   35840 +6144   : Y2_2 packed  (K=16 , N=380, KK=4 , 24 tiles)  TP2
//   41984 +6080   : Bo2T packed  (K=380, N=16 , KK=95,  1 tile )  TP2
// Packed B layout: elem[((nt*KK+kk)*32 + lane)*2 + v] =
//   M[4*kk + 2*(lane>>4) + v][16*nt + (lane&15)]  (zero outside K,N)
// =====================================================================

// ---------------- setup 1: Gauss-Legendre nodes (f64 Newton) ----------
__global__ void tp_gl_nodes(float* wsf) {
  double* wsd = (double*)wsf;
  int tid = threadIdx.x;
  int n = 0, i = -1, xo = 0, wo = 0;
  if (tid < 14)                 { n = 14; i = tid;      xo = 0;  wo = 16; }
  else if (tid >= 32 && tid < 52) { n = 20; i = tid - 32; xo = 40; wo = 64; }
  if (i < 0) return;
  double x = cos(PI_D * (i + 0.75) / (n + 0.5));
  for (int it = 0; it < 64; ++it) {
    double p0 = 1.0, p1 = x;
    for (int k = 2; k <= n; ++k) {
      double p2 = ((2.0 * k - 1.0) * x * p1 - (k - 1.0) * p0) / (double)k;
      p0 = p1; p1 = p2;
    }
    double dp = n * (x * p1 - p0) / (x * x - 1.0);
    x -= p1 / dp;
  }
  double p0 = 1.0, p1 = x;
  for (int k = 2; k <= n; ++k) {
    double p2 = ((2.0 * k - 1.0) * x * p1 - (k - 1.0) * p0) / (double)k;
    p0 = p1; p1 = p2;
  }
  double dp = n * (x * p1 - p0) / (x * x - 1.0);
  wsd[xo + i] = x;
  wsd[wo + i] = 2.0 / ((1.0 - x * x) * dp * dp);
}

// ---------------- setup 2: packed SH matrices -------------------------
// Orthonormal real SH component `comp` at cos(beta)=x, alpha index ai.
__device__ double tp_ylm(int comp, double x, int ai, int ra) {
  int l = 0;
  while ((l + 1) * (l + 1) <= comp) ++l;
  int m  = comp - l * l - l;
  int am = m < 0 ? -m : m;
  double sb = sqrt(fmax(0.0, 1.0 - x * x));
  double p  = sqrt(1.0 / (4.0 * PI_D));                // P[0][0]
  for (int mm = 1; mm <= am; ++mm)
    p = -sqrt((2.0 * mm + 1.0) / (2.0 * mm)) * sb * p; // P[m][m]
  double plm = p;
  if (l > am) {
    double pp = p;
    double pc = sqrt(2.0 * am + 3.0) * x * p;          // P[m+1][m]
    for (int ll = am + 2; ll <= l; ++ll) {
      double a  = sqrt((4.0 * (double)ll * ll - 1.0) /
                       ((double)ll * ll - (double)am * am));
      double bb = sqrt(((double)(ll - 1) * (ll - 1) - (double)am * am) /
                       (4.0 * (double)(ll - 1) * (ll - 1) - 1.0));
      double pn = a * (x * pc - bb * pp);
      pp = pc; pc = pn;
    }
    plm = pc;
  }
  if (m == 0) return plm;
  double alpha = 2.0 * PI_D * (double)ai / (double)ra;
  double s2 = sqrt(2.0);
  return (m > 0) ? s2 * plm * cos((double)m * alpha)
                 : s2 * plm * sin((double)am * alpha);
}

__global__ void tp_pack(float* wsf) {
  const double* wsd = (const double*)wsf;
  int idx = blockIdx.x * blockDim.x + threadIdx.x;
  if (idx >= 47040) return;
  int off, K, N, KK, tp, isbo, rel = idx;
  if (rel < 3072)                  { off = 1024;  K = 16;  N = 182; KK = 4;  tp = 1; isbo = 0; }
  else if ((rel -= 3072) < 11776)  { off = 4096;  K = 182; N = 49;  KK = 46; tp = 1; isbo = 1; }
  else if ((rel -= 11776) < 19968) { off = 15872; K = 49;  N = 380; KK = 13; tp = 2; isbo = 0; }
  else if ((rel -= 19968) < 6144)  { off = 35840; K = 16;  N = 380; KK = 4;  tp = 2; isbo = 0; }
  else { rel -= 6144;                off = 41984; K = 380; N = 16;  KK = 95; tp = 2; isbo = 1; }
  int pos = rel >> 6, b6 = rel & 63;
  int ln = b6 >> 1, v = b6 & 1;
  int nt = pos / KK, kk = pos - nt * KK;
  int k = kk * 4 + (ln >> 4) * 2 + v;
  int n = nt * 16 + (ln & 15);
  double val = 0.0;
  if (k < K && n < N) {
    int ra = (tp == 1) ? 13 : 19;
    const double* xs = wsd + ((tp == 1) ? 0 : 40);
    const double* wg = wsd + ((tp == 1) ? 16 : 64);
    int comp = isbo ? n : k;
    int g    = isbo ? k : n;
    int bn = g / ra, ai = g - bn * ra;
    val = tp_ylm(comp, xs[bn], ai, ra);
    if (isbo) val *= wg[bn] * (2.0 * PI_D / (double)ra);
  }
  wsf[off + rel] = (float)val;
}

// ---------------- main: WMMA pipeline ---------------------------------
__device__ __forceinline__ v8f wmma4(v2f a, v2f b, v8f c) {
  return __builtin_amdgcn_wmma_f32_16x16x4_f32(false, a, false, b,
                                               (short)0, c, false, false);
}

// Per-wave LDS (floats): F 16x18 | wtab 192 | P2 16x66 | G 16x386
#define LW 7712

__global__ __launch_bounds__(64) void tp_main(
    const float* __restrict__ feat, const float* __restrict__ atype,
    const float* __restrict__ w1, const float* __restrict__ w2,
    const float* __restrict__ w3, const float* __restrict__ wsf,
    float* __restrict__ out)
{
  __shared__ float lds[2 * LW];
  const int lane = threadIdx.x & 31;
  const int wave = threadIdx.x >> 5;
  float* Lb = lds + wave * LW;
  float* Fb = Lb;          // 16 rows, stride 18
  float* WT = Lb + 288;    // 16 x (3 nu x 4 l)
  float* Pb = Lb + 480;    // 16 rows, stride 66
  float* Gb = Lb + 1536;   // 16 rows, stride 194 (TP1) / 386 (TP2)

  const int tile = blockIdx.x * 2 + wave;     // 8192 tiles of 16 rows
  const int m0 = tile << 4;
  const int b  = m0 >> 7;                     // C=128 divides tile rows
  const int c0 = m0 & 127;
  const int hi = lane >> 4;
  const int nl = lane & 15;

  // --- stage F tile into LDS (row-major, padded stride) ---
  const float* fr = feat + (size_t)m0 * 16;
#pragma unroll
  for (int j = 0; j < 8; ++j) {
    int e = lane * 8 + j;
    Fb[(e >> 4) * 18 + (e & 15)] = fr[e];
  }

  // --- per-row atom-type-contracted weights: WT[r][nu][l] ---
  const float* at = atype + b * 10;
#pragma unroll
  for (int t6 = 0; t6 < 6; ++t6) {
    int t = t6 * 32 + lane;                   // 0..191
    int r = t / 12, rem = t - r * 12, nu = rem >> 2, l = rem & 3;
    const float* wp = (nu == 0 ? w1 : (nu == 1 ? w2 : w3)) + ((c0 + r) << 2) + l;
    float a = 0.f;
#pragma unroll
    for (int e = 0; e < 10; ++e) a += at[e] * wp[e * 512];
    WT[t] = a;
  }

  const float* PY1  = wsf + 1024;
  const float* PB1  = wsf + 4096;
  const float* PY12 = wsf + 15872;
  const float* PY22 = wsf + 35840;
  const float* PB2  = wsf + 41984;
  const v8f vz = {0.f, 0.f, 0.f, 0.f, 0.f, 0.f, 0.f, 0.f};

  // --- TP1 to-grid + square:  G = (F·Y1) ⊙ (F·Y1),  182 cols (12 tiles)
  for (int nt = 0; nt < 12; ++nt) {
    v8f acc = vz;
#pragma unroll
    for (int kk = 0; kk < 4; ++kk)
      acc = wmma4(*(const v2f*)(Fb + nl * 18 + kk * 4 + hi * 2),
                  *(const v2f*)(PY1 + (nt * 4 + kk) * 64 + lane * 2), acc);
    acc = acc * acc;                          // s1 == s2 for nu=1
#pragma unroll
    for (int r = 0; r < 8; ++r)
      Gb[(r + 8 * hi) * 194 + nt * 16 + nl] = acc[r];
  }

  // --- from-grid:  P2 = G · Bo1ᵀ  (K pad 184, N pad 64; pads are 0)
  for (int nt = 0; nt < 4; ++nt) {
    v8f acc = vz;
#pragma unroll 2
    for (int kk = 0; kk < 46; ++kk)
      acc = wmma4(*(const v2f*)(Gb + nl * 194 + kk * 4 + hi * 2),
                  *(const v2f*)(PB1 + (nt * 46 + kk) * 64 + lane * 2), acc);
#pragma unroll
    for (int r = 0; r < 8; ++r)
      Pb[(r + 8 * hi) * 66 + nt * 16 + nl] = acc[r];
  }

  // --- TP2 to-grid + product:  G2 = (P2·Y1₂) ⊙ (F·Y2₂), 380 cols (24 tiles)
  for (int nt = 0; nt < 24; ++nt) {
    v8f a1 = vz, a2 = vz;
#pragma unroll 2
    for (int kk = 0; kk < 13; ++kk)
      a1 = wmma4(*(const v2f*)(Pb + nl * 66 + kk * 4 + hi * 2),
                 *(const v2f*)(PY12 + (nt * 13 + kk) * 64 + lane * 2), a1);
#pragma unroll
    for (int kk = 0; kk < 4; ++kk)
      a2 = wmma4(*(const v2f*)(Fb + nl * 18 + kk * 4 + hi * 2),
                 *(const v2f*)(PY22 + (nt * 4 + kk) * 64 + lane * 2), a2);
    a1 = a1 * a2;
#pragma unroll
    for (int r = 0; r < 8; ++r)
      Gb[(r + 8 * hi) * 386 + nt * 16 + nl] = a1[r];
  }

  // --- from-grid (only 16 needed coeffs):  P3 = G2 · Bo2ᵀ[:, :16]
  v8f a3 = vz;
#pragma unroll 5
  for (int kk = 0; kk < 95; ++kk)
    a3 = wmma4(*(const v2f*)(Gb + nl * 386 + kk * 4 + hi * 2),
               *(const v2f*)(PB2 + kk * 64 + lane * 2), a3);

  // --- combine nu=1..3 with per-l weights; scatter in split/cat layout
#pragma unroll
  for (int r = 0; r < 8; ++r) {
    int M = r + 8 * hi;
    int N = nl;
    int l = (N >= 9) ? 3 : (N >= 4) ? 2 : (N >= 1) ? 1 : 0;
    int offl = (l == 0) ? 0 : (l == 1) ? 128 : (l == 2) ? 512 : 1152;
    float res = WT[M * 12 + l]     * Fb[M * 18 + N]
              + WT[M * 12 + 4 + l] * Pb[M * 66 + N]
              + WT[M * 12 + 8 + l] * a3[r];
    out[(size_t)b * 2048 + offl + (size_t)(c0 + M) * (2 * l + 1) + (N - l * l)] = res;
  }
}

// ---------------------------------------------------------------------
extern "C" void kernel_launch(void* const* d_in, const int* in_sizes, int n_in,
                              void* d_out, int out_size, void* d_ws, size_t ws_size,
                              hipStream_t stream) {
  (void)in_sizes; (void)n_in; (void)out_size; (void)ws_size;
  const float* feat  = (const float*)d_in[0];
  const float* atype = (const float*)d_in[1];
  const float* w1    = (const float*)d_in[2];
  const float* w2    = (const float*)d_in[3];
  const float* w3    = (const float*)d_in[4];
  float* out = (float*)d_out;
  float* wsf = (float*)d_ws;

  tp_gl_nodes<<<1, 64, 0, stream>>>(wsf);
  tp_pack<<<(47040 + 255) / 256, 256, 0, stream>>>(wsf);
  // 131072 rows / 16 per wave / 2 waves per block
  tp_main<<<4096, 64, 0, stream>>>(feat, atype, w1, w2, w3, wsf, out);
}